// LightGCN_37031208026373
// MI455X (gfx1250) — compile-verified
//
#include <hip/hip_runtime.h>

#define NUM_USERS 60000
#define NUM_ITEMS 30000
#define NN (NUM_USERS + NUM_ITEMS)   // 90000
#define DIM 64
#define NUM_LAYERS 3
#define CHUNK 64                     // edges per async-staged chunk (per wave)

typedef __attribute__((ext_vector_type(2))) float v2f;
typedef __attribute__((ext_vector_type(8))) float v8f;

#if __has_builtin(__builtin_amdgcn_global_load_async_to_lds_b64) && \
    __has_builtin(__builtin_amdgcn_s_wait_asynccnt)
#define LGCN_USE_ASYNC 1
// b64 async payload type, per compiler diagnostic: vector_size(8) int,
// global source in AS(1), LDS destination in AS(3).
typedef int v2i __attribute__((vector_size(8)));
typedef __attribute__((address_space(1))) v2i* gptr_b64_t;
typedef __attribute__((address_space(3))) v2i* lptr_b64_t;
#endif

// ---- init: emb_cur = acc = concat(user,item); emb_next = 0; deg = 0 ----
__global__ void lgcn_init(const float* __restrict__ user_emb,
                          const float* __restrict__ item_emb,
                          float* __restrict__ emb_cur,
                          float* __restrict__ emb_next,
                          float* __restrict__ acc,
                          float* __restrict__ deg) {
  const long long total = (long long)NN * DIM;
  const long long ub = (long long)NUM_USERS * DIM;
  for (long long i = (long long)blockIdx.x * blockDim.x + threadIdx.x; i < total;
       i += (long long)gridDim.x * blockDim.x) {
    float v = (i < ub) ? user_emb[i] : item_emb[i - ub];
    emb_cur[i]  = v;
    acc[i]      = v;
    emb_next[i] = 0.0f;
  }
  for (int i = blockIdx.x * blockDim.x + threadIdx.x; i < NN;
       i += gridDim.x * blockDim.x)
    deg[i] = 0.0f;
}

// ---- degree counting over symmetrized edges ----
__global__ void lgcn_degree(const int* __restrict__ rows,
                            const int* __restrict__ cols,
                            int E, float* __restrict__ deg) {
  for (int e = blockIdx.x * blockDim.x + threadIdx.x; e < E;
       e += gridDim.x * blockDim.x) {
    unsafeAtomicAdd(&deg[rows[e]], 1.0f);
    unsafeAtomicAdd(&deg[cols[e]], 1.0f);
  }
}

// ---- deg -> deg^{-1/2} in place ----
__global__ void lgcn_rsqrt(float* __restrict__ deg) {
  for (int i = blockIdx.x * blockDim.x + threadIdx.x; i < NN;
       i += gridDim.x * blockDim.x) {
    float d = deg[i];
    deg[i] = (d > 0.0f) ? rsqrtf(d) : 0.0f;
  }
}

// ---- per-edge work: wave32 handles both directions, 2 dims per lane ----
__device__ __forceinline__ void lgcn_process_edge(int u, int it,
                                                  const float* __restrict__ dinv,
                                                  const v2f* __restrict__ src2,
                                                  float* __restrict__ dst,
                                                  int lane) {
  const float w = dinv[u] * dinv[it];
  const v2f su = src2[u  * (DIM / 2) + lane];      // global_load_b64
  const v2f si = src2[it * (DIM / 2) + lane];
  float* du = dst + (long long)u  * DIM + lane * 2;
  float* di = dst + (long long)it * DIM + lane * 2;
  unsafeAtomicAdd(di + 0, w * su.x);               // global_atomic_add_f32 (L2)
  unsafeAtomicAdd(di + 1, w * su.y);
  unsafeAtomicAdd(du + 0, w * si.x);
  unsafeAtomicAdd(du + 1, w * si.y);
}

// ---- one propagation layer.
// Async path: each wave DMAs CHUNK edge ids (rows+cols) into a wave-private
// LDS slice with GLOBAL_LOAD_ASYNC_TO_LDS_B64 (double-buffered, ASYNCcnt),
// then reads indices via broadcast ds_load_b32. Tail edges go direct.
__global__ void lgcn_scatter(const int* __restrict__ rows,
                             const int* __restrict__ cols,
                             const float* __restrict__ dinv,
                             const float* __restrict__ src,
                             float* __restrict__ dst, int E, int fullChunks) {
  const int lane   = threadIdx.x & 31;
  const int gwave  = (blockIdx.x * blockDim.x + threadIdx.x) >> 5;
  const int nwaves = (gridDim.x * blockDim.x) >> 5;
  const v2f* __restrict__ src2 = (const v2f*)src;

#if defined(LGCN_USE_ASYNC)
  const int wv = threadIdx.x >> 5;                 // wave in block, 0..7
  // [wave][buf][rows(CHUNK) | cols(CHUNK)]
  __shared__ __align__(16) int lds[8][2][2 * CHUNK];

  // Issue one chunk's async copies: 2 x b64 per wave (each lane moves 8B/array).
  auto issue = [&](int c, int b) {
    const int* gr = rows + c * CHUNK + lane * 2;
    const int* gc = cols + c * CHUNK + lane * 2;
    // LDS byte address = low 32 bits of the flat address (ISA aperture rule).
    lptr_b64_t lr = (lptr_b64_t)(unsigned int)(unsigned long long)
                        (const void*)&lds[wv][b][lane * 2];
    lptr_b64_t lc = (lptr_b64_t)(unsigned int)(unsigned long long)
                        (const void*)&lds[wv][b][CHUNK + lane * 2];
    __builtin_amdgcn_global_load_async_to_lds_b64(
        (gptr_b64_t)(unsigned long long)gr, lr, 0, 0);
    __builtin_amdgcn_global_load_async_to_lds_b64(
        (gptr_b64_t)(unsigned long long)gc, lc, 0, 0);
  };

  int buf = 0;
  if (gwave < fullChunks) issue(gwave, buf);
  for (int c = gwave; c < fullChunks; c += nwaves) {
    const int nc = c + nwaves;
    if (nc < fullChunks) {
      issue(nc, buf ^ 1);
      __builtin_amdgcn_s_wait_asynccnt(2);         // current chunk landed
    } else {
      __builtin_amdgcn_s_wait_asynccnt(0);
    }
#pragma unroll 1
    for (int j = 0; j < CHUNK; ++j) {
      const int u  = lds[wv][buf][j];              // ds_load_b32 broadcast
      const int it = lds[wv][buf][CHUNK + j];
      lgcn_process_edge(u, it, dinv, src2, dst, lane);
    }
    buf ^= 1;
  }
  // tail: E % CHUNK edges, direct path
  for (int e = fullChunks * CHUNK + gwave; e < E; e += nwaves)
    lgcn_process_edge(rows[e], cols[e], dinv, src2, dst, lane);
#else
  for (int e = gwave; e < E; e += nwaves) {
    const int u  = rows[e];
    const int it = cols[e];
    if (lane == 0 && (e + nwaves) < E) {
      __builtin_prefetch(&rows[e + nwaves], 0, 0); // global_prefetch_b8
      __builtin_prefetch(&cols[e + nwaves], 0, 0);
    }
    lgcn_process_edge(u, it, dinv, src2, dst, lane);
  }
#endif
}

// ---- acc += layer output; zero the buffer that becomes next layer's target ----
__global__ void lgcn_accum_zero(const float* __restrict__ nxt,
                                float* __restrict__ acc,
                                float* __restrict__ old_cur) {
  const long long total = (long long)NN * DIM;
  for (long long i = (long long)blockIdx.x * blockDim.x + threadIdx.x; i < total;
       i += (long long)gridDim.x * blockDim.x) {
    acc[i] += nxt[i];
    old_cur[i] = 0.0f;
  }
}

// ---- epilogue: acc *= 0.25 on the matrix engine, D = (0.25*I16) x X per tile,
// decomposed into 4 chained V_WMMA_F32_16X16X4_F32 (exact in fp32).
// A (16x4): lane m=lane&15, h=lane>>4; VGPR{0,1} hold K={2h,2h+1}.
// B (4x16):  lane n=lane&15;           VGPR{0,1} hold K={2h,2h+1}.
// D (16x16): VGPR v holds rows {v, v+8}, column = lane&15.
__global__ void lgcn_scale_wmma(float* __restrict__ acc) {
  const int lane   = threadIdx.x & 31;
  const int wave   = (blockIdx.x * blockDim.x + threadIdx.x) >> 5;
  const int nwaves = (gridDim.x * blockDim.x) >> 5;
  const int tilesPerRow = DIM / 16;                // 4
  const int ntiles = (NN / 16) * tilesPerRow;      // 22500, exact
  const int h = lane >> 4;
  const int n = lane & 15;
  for (int t = wave; t < ntiles; t += nwaves) {
    const int tr = t / tilesPerRow;
    const int tc = t - tr * tilesPerRow;
    float* base = acc + (long long)tr * 16 * DIM + tc * 16;
    v8f c = {};
#pragma unroll
    for (int k = 0; k < 4; ++k) {
      const int r0 = 4 * k + 2 * h;
      v2f a, b;
      a.x = (n == r0    ) ? 0.25f : 0.0f;
      a.y = (n == r0 + 1) ? 0.25f : 0.0f;
      b.x = base[(r0    ) * DIM + n];
      b.y = base[(r0 + 1) * DIM + n];
      c = __builtin_amdgcn_wmma_f32_16x16x4_f32(
              /*neg_a=*/false, a, /*neg_b=*/false, b,
              /*c_mod=*/(short)0, c, /*reuse_a=*/false, /*reuse_b=*/false);
    }
#pragma unroll
    for (int v = 0; v < 8; ++v)
      base[(v + 8 * h) * DIM + n] = c[v];
  }
}

extern "C" void kernel_launch(void* const* d_in, const int* in_sizes, int n_in,
                              void* d_out, int out_size, void* d_ws, size_t ws_size,
                              hipStream_t stream) {
  const int*   edge     = (const int*)d_in[0];     // [2, E] flat: rows then cols
  const float* user_emb = (const float*)d_in[1];
  const float* item_emb = (const float*)d_in[2];
  const int E = in_sizes[0] / 2;
  const int* rows = edge;
  const int* cols = edge + E;
  const int fullChunks = E / CHUNK;

  float* acc = (float*)d_out;                      // [NN, 64] accumulator/output

  // workspace: deg/dinv (NN) | emb_a (NN*64) | emb_b (NN*64)  ~= 46.4 MB
  float* deg   = (float*)d_ws;
  float* emb_a = deg + NN;
  float* emb_b = emb_a + (size_t)NN * DIM;

  lgcn_init<<<4096, 256, 0, stream>>>(user_emb, item_emb, emb_a, emb_b, acc, deg);
  lgcn_degree<<<2048, 256, 0, stream>>>(rows, cols, E, deg);
  lgcn_rsqrt<<<(NN + 255) / 256, 256, 0, stream>>>(deg);

  float* cur = emb_a;
  float* nxt = emb_b;
  for (int l = 0; l < NUM_LAYERS; ++l) {
    lgcn_scatter<<<1024, 256, 0, stream>>>(rows, cols, deg, cur, nxt, E, fullChunks);
    lgcn_accum_zero<<<4096, 256, 0, stream>>>(nxt, acc, cur);
    float* tmp = cur; cur = nxt; nxt = tmp;        // host-side swap, deterministic
  }
  lgcn_scale_wmma<<<2048, 256, 0, stream>>>(acc);
}